// WTA_layer_Neuron_80891414053662
// MI455X (gfx1250) — compile-verified
//
#include <hip/hip_runtime.h>
#include <cstdint>

// WTA spiking layer temporal scan for MI455X (gfx1250).
// Memory-bound streaming scan: async global->LDS double buffering (ASYNCcnt),
// non-temporal cache policy (read-once / write-once streams), wave32 shuffle
// max-reduction, all neuron state carried in VGPRs for the whole scan.

typedef float v4f __attribute__((ext_vector_type(4))); // native vector for NT store

constexpr int T_STEPS = 36;   // START + TIMESTEP + WAIT = 0 + 24 + 12
constexpr int FIRE_T  = 12;   // START + WAIT
constexpr int N       = 1024; // neurons per row
constexpr int THREADS = 256;  // 8 wave32 per row
constexpr int PER_THR = N / THREADS; // 4 elements per thread

// V_TH0 = tau^(TIMESTEP-1) / (1/ln tau) / 2 = 2^23 * ln(2) / 2 = 2^22 * ln 2
__device__ __constant__ float kThInit = (float)(4194304.0 * 0.69314718055994530942);

__global__ __launch_bounds__(THREADS)
void wta_scan_kernel(const float* __restrict__ x, float* __restrict__ out, int rows)
{
    const int row  = blockIdx.x;
    const int tid  = threadIdx.x;
    const int lane = tid & 31;
    const int wid  = tid >> 5;

    __shared__ __align__(16) float smem[2][N]; // double-buffered input row
    __shared__ float red[2][8];                // parity-doubled max partials

    const size_t rowoff  = (size_t)row * N + (size_t)tid * PER_THR;
    const size_t tstride = (size_t)rows * N;

    // ---- per-element carried state (registers for the whole scan) ----
    float v[PER_THR];
    #pragma unroll
    for (int i = 0; i < PER_THR; ++i) v[i] = 0.0f;
    unsigned blocked = 0u;        // 4-bit "already spiked" mask
    float block_input = 1.0f;     // per-row uniform, updated identically
    float th = kThInit;           // deterministic scalar threshold

    // ---- prologue: async-stage timestep 0 into LDS buffer 0 ----
    {
        unsigned lds = (unsigned)(uintptr_t)&smem[0][tid * PER_THR];
        unsigned long long ga = (unsigned long long)(x + rowoff);
        asm volatile("global_load_async_to_lds_b128 %0, %1, off th:TH_LOAD_NT"
                     :: "v"(lds), "v"(ga) : "memory");
    }

    for (int t = 0; t < T_STEPS; ++t) {
        const int cur = t & 1;

        if (t + 1 < T_STEPS) {
            // ensure this wave's prior LDS reads of the recycled buffer retired
            asm volatile("s_wait_dscnt 0x0" ::: "memory");
            unsigned lds = (unsigned)(uintptr_t)&smem[(t + 1) & 1][tid * PER_THR];
            unsigned long long ga =
                (unsigned long long)(x + (size_t)(t + 1) * tstride + rowoff);
            asm volatile("global_load_async_to_lds_b128 %0, %1, off th:TH_LOAD_NT"
                         :: "v"(lds), "v"(ga) : "memory");
            // async loads complete in order: <=1 outstanding => step t's data landed
            asm volatile("s_wait_asynccnt 0x1" ::: "memory");
        } else {
            asm volatile("s_wait_asynccnt 0x0" ::: "memory");
        }

        const v4f xv4 = *reinterpret_cast<const v4f*>(&smem[cur][tid * PER_THR]);
        float xv[PER_THR] = {xv4.x, xv4.y, xv4.z, xv4.w};

        // ---- row max of raw input (WTA input mask) ----
        float m = fmaxf(fmaxf(xv[0], xv[1]), fmaxf(xv[2], xv[3]));
        #pragma unroll
        for (int off = 16; off > 0; off >>= 1)
            m = fmaxf(m, __shfl_xor(m, off, 32));
        if (lane == 0) red[cur][wid] = m;
        __syncthreads();                 // single barrier: parity buffer makes the
        float tp = red[cur][0];          // trailing barrier provably redundant
        #pragma unroll
        for (int i = 1; i < 8; ++i) tp = fmaxf(tp, red[cur][i]);

        // ---- integrate (input-side WTA subtraction) ----
        const float sub = tp * block_input;
        #pragma unroll
        for (int i = 0; i < PER_THR; ++i) v[i] += xv[i] - sub;
        if (tp > 0.0f) block_input = 0.0f;

        // ---- fire / soft reset / output-side blocking ----
        float o[PER_THR] = {0.0f, 0.0f, 0.0f, 0.0f};
        if (t >= FIRE_T) {
            #pragma unroll
            for (int i = 0; i < PER_THR; ++i) {
                const float spike = (v[i] >= th) ? 1.0f : 0.0f;
                v[i] -= spike * th;                       // soft reset (always)
                const float oi = (blocked & (1u << i)) ? 0.0f : spike;
                o[i] = oi;
                if (oi > 0.0f) blocked |= (1u << i);      // each neuron fires once
            }
            th *= 0.5f;                                   // th /= tau
        }

        // write-once stream: non-temporal store (th:TH_NT), don't dirty L2
        v4f ov = {o[0], o[1], o[2], o[3]};
        __builtin_nontemporal_store(
            ov, reinterpret_cast<v4f*>(out + (size_t)t * tstride + rowoff));
    }
}

extern "C" void kernel_launch(void* const* d_in, const int* in_sizes, int n_in,
                              void* d_out, int out_size, void* d_ws, size_t ws_size,
                              hipStream_t stream)
{
    const float* x = (const float*)d_in[0];
    float* out = (float*)d_out;
    const int total = in_sizes[0];          // T * B * H * N
    const int rows = total / (T_STEPS * N); // 32 * 64 = 2048
    wta_scan_kernel<<<dim3(rows), dim3(THREADS), 0, stream>>>(x, out, rows);
}